// CornerNet_87531433492875
// MI455X (gfx1250) — compile-verified
//
#include <hip/hip_runtime.h>
#include <hip/hip_bf16.h>
#include <math.h>

// Problem sizes (fixed by the reference)
#define B_TOT 2048
#define R_TOT 512
#define D_TOT 256
#define BT 64          // b-tile per workgroup
#define RT 8           // r per workgroup (one per wave)
#define D4 (D_TOT / 4) // 64 float4 chunks over d
#define RD (R_TOT * D_TOT)
#define NRB (R_TOT / RT) // 64 r-blocks

typedef __attribute__((ext_vector_type(2))) float v2f;
typedef __attribute__((ext_vector_type(8))) float v8f;

__device__ __forceinline__ float fast_exp2(float x) { return __builtin_amdgcn_exp2f(x); }
__device__ __forceinline__ float rcp_nr(float d) {
  float r = __builtin_amdgcn_rcpf(d);
  return r * (2.0f - d * r); // one Newton step: ~0.5 ulp
}

#define LOG2E 1.4426950408889634f

// ---------------------------------------------------------------------------
// Kernel 0: fold kappa/tanh/sigmoid into per-(r,d) constants.
//   negA[i] = -kappa*tanh(sign)*log2e
//   Cc[i]   =  kappa*tanh(sign)*log2e * th
//   om[i]   =  1 - sigmoid(mask_logit)
// Then per element: E = 2^(negA*x + Cc); gated = (1 + om*E) / (1 + E).
// ---------------------------------------------------------------------------
__global__ void cn_precompute(const float* __restrict__ th,
                              const float* __restrict__ sign_param,
                              const float* __restrict__ mask_logit,
                              const float* __restrict__ log_kappa,
                              float* __restrict__ negA,
                              float* __restrict__ Cc,
                              float* __restrict__ om) {
  int idx = blockIdx.x * blockDim.x + threadIdx.x;
  if (idx >= RD) return;
  float kappa = expf(log_kappa[0]);
  float a = kappa * tanhf(sign_param[idx]) * LOG2E;
  negA[idx] = -a;
  Cc[idx] = a * th[idx];
  float m = 1.0f / (1.0f + expf(-mask_logit[idx]));
  om[idx] = 1.0f - m;
}

__global__ void cn_out_init(const float* __restrict__ head_b, float* __restrict__ out) {
  int b = blockIdx.x * blockDim.x + threadIdx.x;
  if (b < B_TOT) out[b] = head_b[0];
}

// ---------------------------------------------------------------------------
// Main kernel. Grid = (B/64, R/8), block = 256 (8 wave32).
// Wave w handles r = r0 + w; lane L handles tile-b {L, L+32}.
// x tile staged in 64KB LDS, XOR-swizzled at float4 granularity.
// PRE=true : params from workspace, deterministic partial-y output per r-block.
// PRE=false: params recomputed inline, atomic accumulation into y.
// ---------------------------------------------------------------------------
template <bool PRE>
__global__ __launch_bounds__(256)
void cn_main(const float* __restrict__ x,
             const float* __restrict__ p0,   // negA  | sign_param
             const float* __restrict__ p1,   // Cc    | th
             const float* __restrict__ p2,   // om    | mask_logit
             const float* __restrict__ log_kappa,
             const float* __restrict__ head_w,
             float* __restrict__ yout) {     // ypart[NRB][B] | y[B]
  __shared__ float4 smem4[BT * D4]; // 64 KB, reused for z-tile afterwards

  const int tid = threadIdx.x;
  const int lane = tid & 31;
  const int wave = tid >> 5;
  const int b0 = blockIdx.x * BT;
  const int r0 = blockIdx.y * RT;

  // ---- Stage x tile (coalesced float4 loads, XOR-swizzled LDS writes) ----
#pragma unroll
  for (int i = 0; i < 16; ++i) {
    int linear = i * 256 + tid;     // 0..4095 float4 slots
    int b = linear >> 6;            // 0..63
    int c4 = linear & 63;           // 0..63
    float4 v = *reinterpret_cast<const float4*>(x + (size_t)(b0 + b) * D_TOT + c4 * 4);
    smem4[b * 64 + (c4 ^ b)] = v;
  }
  __syncthreads();

  const int r = __builtin_amdgcn_readfirstlane(r0 + wave); // wave-uniform
  const float4* q0 = reinterpret_cast<const float4*>(p0) + (size_t)r * D4;
  const float4* q1 = reinterpret_cast<const float4*>(p1) + (size_t)r * D4;
  const float4* q2 = reinterpret_cast<const float4*>(p2) + (size_t)r * D4;

  float kL2E = 0.0f;
  if (!PRE) kL2E = fast_exp2(log_kappa[0] * LOG2E) * LOG2E; // kappa*log2e

  const int bA = lane, bB = lane + 32;
  float prod0 = 1.0f, prod1 = 1.0f;

#define ELEM(na_, cc_, om_, xv_, num, den)      \
  {                                             \
    float u = fmaf(na_, xv_, cc_);              \
    u = fminf(u, 30.0f);                        \
    float E = fast_exp2(u);                     \
    num *= fmaf(om_, E, 1.0f);                  \
    den = fmaf(den, E, den);                    \
  }

  for (int d4 = 0; d4 < D4; ++d4) {
    float4 na, cc, omv;
    if (PRE) {
      na = q0[d4]; cc = q1[d4]; omv = q2[d4];
    } else {
      float4 sp = q0[d4], thv = q1[d4], ml = q2[d4];
      float ax = kL2E * tanhf(sp.x), ay = kL2E * tanhf(sp.y);
      float az = kL2E * tanhf(sp.z), aw = kL2E * tanhf(sp.w);
      na.x = -ax; na.y = -ay; na.z = -az; na.w = -aw;
      cc.x = ax * thv.x; cc.y = ay * thv.y; cc.z = az * thv.z; cc.w = aw * thv.w;
      omv.x = 1.0f - rcp_nr(1.0f + fast_exp2(-ml.x * LOG2E));
      omv.y = 1.0f - rcp_nr(1.0f + fast_exp2(-ml.y * LOG2E));
      omv.z = 1.0f - rcp_nr(1.0f + fast_exp2(-ml.z * LOG2E));
      omv.w = 1.0f - rcp_nr(1.0f + fast_exp2(-ml.w * LOG2E));
    }
    float4 xv0 = smem4[bA * 64 + (d4 ^ bA)];
    float4 xv1 = smem4[bB * 64 + (d4 ^ bB)];

    float num0 = 1.0f, den0 = 1.0f, num1 = 1.0f, den1 = 1.0f;
    ELEM(na.x, cc.x, omv.x, xv0.x, num0, den0);
    ELEM(na.y, cc.y, omv.y, xv0.y, num0, den0);
    ELEM(na.z, cc.z, omv.z, xv0.z, num0, den0);
    ELEM(na.w, cc.w, omv.w, xv0.w, num0, den0);
    ELEM(na.x, cc.x, omv.x, xv1.x, num1, den1);
    ELEM(na.y, cc.y, omv.y, xv1.y, num1, den1);
    ELEM(na.z, cc.z, omv.z, xv1.z, num1, den1);
    ELEM(na.w, cc.w, omv.w, xv1.w, num1, den1);
    prod0 *= num0 * rcp_nr(den0);
    prod1 *= num1 * rcp_nr(den1);
  }
#undef ELEM

  // ---- z tile -> LDS (reuse x buffer), then WMMA head reduction ----
  __syncthreads(); // all waves done reading x
  float* zsm = reinterpret_cast<float*>(smem4); // [64][RT]
  zsm[bA * RT + wave] = prod0;
  zsm[bB * RT + wave] = prod1;
  __syncthreads();

  if (wave < 4) {
    // Wave bg reduces b rows [bg*16, bg*16+16) over the 8 r of this block:
    // D = A2*B + (A1*B + 0), A[m][k] = z[m][k]*w[k], B = all-ones (layout-free).
    // A 16x4 f32 layout (ISA 7.12.2): lanes 0-15 -> M=lane, K={0,1};
    //                                 lanes 16-31 -> M=lane-16, K={2,3}.
    const int bg = wave;
    const int M = lane & 15;
    const int kb = (lane >> 4) << 1; // 0 or 2
    const float* zrow = zsm + (bg * 16 + M) * RT;
    const float w0 = head_w[r0 + kb];
    const float w1 = head_w[r0 + kb + 1];
    const float w2 = head_w[r0 + 4 + kb];
    const float w3 = head_w[r0 + 4 + kb + 1];
    v2f a0, a1, bones;
    a0.x = zrow[kb] * w0;     a0.y = zrow[kb + 1] * w1;
    a1.x = zrow[4 + kb] * w2; a1.y = zrow[4 + kb + 1] * w3;
    bones.x = 1.0f; bones.y = 1.0f;
    v8f acc = {};
    acc = __builtin_amdgcn_wmma_f32_16x16x4_f32(false, a0, false, bones,
                                                (short)0, acc, false, false);
    acc = __builtin_amdgcn_wmma_f32_16x16x4_f32(false, a1, false, bones,
                                                (short)0, acc, false, false);
    // Column N=0 lives in lane 0 (M=0..7 over v0..v7) and lane 16 (M=8..15).
    if ((lane & 15) == 0) {
      int brow = b0 + bg * 16 + ((lane >> 4) << 3);
      if (PRE) {
        float* yp = yout + (size_t)blockIdx.y * B_TOT + brow;
#pragma unroll
        for (int i = 0; i < 8; ++i) yp[i] = acc[i]; // deterministic partials
      } else {
#pragma unroll
        for (int i = 0; i < 8; ++i) atomicAdd(&yout[brow + i], acc[i]);
      }
    }
  }
}

// Final reduction over r-blocks (deterministic path)
__global__ void cn_reduce(const float* __restrict__ ypart,
                          const float* __restrict__ head_b,
                          float* __restrict__ out) {
  int b = blockIdx.x * blockDim.x + threadIdx.x;
  if (b >= B_TOT) return;
  float s = head_b[0];
#pragma unroll 8
  for (int j = 0; j < NRB; ++j) s += ypart[(size_t)j * B_TOT + b];
  out[b] = s;
}

// ---------------------------------------------------------------------------
extern "C" void kernel_launch(void* const* d_in, const int* in_sizes, int n_in,
                              void* d_out, int out_size, void* d_ws, size_t ws_size,
                              hipStream_t stream) {
  (void)in_sizes; (void)n_in; (void)out_size;
  const float* x  = (const float*)d_in[0];
  const float* th = (const float*)d_in[1];
  const float* sp = (const float*)d_in[2];
  const float* ml = (const float*)d_in[3];
  const float* lk = (const float*)d_in[4];
  const float* hw = (const float*)d_in[5];
  const float* hb = (const float*)d_in[6];
  float* out = (float*)d_out;

  const size_t paramBytes = (size_t)3 * RD * sizeof(float);          // 1.5 MB
  const size_t ypartBytes = (size_t)NRB * B_TOT * sizeof(float);     // 0.5 MB
  dim3 grid(B_TOT / BT, R_TOT / RT);

  if (d_ws && ws_size >= paramBytes + ypartBytes) {
    float* negA = (float*)d_ws;
    float* Cc = negA + RD;
    float* om = Cc + RD;
    float* ypart = om + RD;
    cn_precompute<<<(RD + 255) / 256, 256, 0, stream>>>(th, sp, ml, lk, negA, Cc, om);
    cn_main<true><<<grid, 256, 0, stream>>>(x, negA, Cc, om, lk, hw, ypart);
    cn_reduce<<<(B_TOT + 255) / 256, 256, 0, stream>>>(ypart, hb, out);
  } else {
    cn_out_init<<<(B_TOT + 255) / 256, 256, 0, stream>>>(hb, out);
    cn_main<false><<<grid, 256, 0, stream>>>(x, sp, th, ml, lk, hw, out);
  }
}